// PixelLossWPT_85134841741509
// MI455X (gfx1250) — compile-verified
//
#include <hip/hip_runtime.h>

typedef __attribute__((ext_vector_type(2))) float v2f;
typedef __attribute__((ext_vector_type(8))) float v8f;

#define TH 16      // output tile rows per block
#define TW 64      // output tile cols per block (4 groups of 16)
#define TR 34      // input tile rows  = 2*TH + 2
#define TC 130     // input tile cols  = 2*TW + 2
#define TSTR 132   // LDS row stride (even -> 8B-aligned v2f loads)

// db2 decomposition filters, flipped (ptwt cross-correlates flipped pywt filters)
#define LO0  0.4829629131445341f
#define LO1  0.8365163037378079f
#define LO2  0.22414386804185735f
#define LO3 -0.12940952255092145f
#define HI0 -0.12940952255092145f
#define HI1 -0.22414386804185735f
#define HI2  0.8365163037378079f
#define HI3 -0.4829629131445341f

__device__ __forceinline__ int refl(int i, int n) {
  if (i < 0)  i = -i;            // jnp 'reflect': -1 -> 1, -2 -> 2
  if (i >= n) i = 2 * n - 2 - i; // H -> H-2, H+1 -> H-3
  if (i < 0)  i = 0;             // clamp for out-of-tile garbage lanes
  if (i >= n) i = n - 1;
  return i;
}

// One 2D db2 DWT step, direct 4x4 conv, stride 2, via V_WMMA_F32_16X16X4_F32.
// MODE 0: input = inA - inB (level 1 diff fusion), store 4 bands
// MODE 1: input = inA, store 4 bands
// MODE 2: input = inA, fuse per-leaf sum-of-squares into gacc (level 3)
// Wsi / Wso: padded row strides (floats) of input / output buffers.
// Wso is a multiple of 8 >= Wp, so a store group starting at jb < Wp may always
// write all 8 floats (tail lands in pad columns that are never read back).
template <int MODE>
__global__ __launch_bounds__(256) void dwt_kernel(
    const float* __restrict__ inA, const float* __restrict__ inB,
    float* __restrict__ out, float* __restrict__ gacc,
    int H, int W, int Wsi, int Hp, int Wp, int Wso, int nb, int tilesC) {
  __shared__ float tile[TR * TSTR];
  __shared__ float lacc[4];

  const int tid = threadIdx.x;
  const int gy  = blockIdx.y;                 // plane = img_in_chunk * nb + cin
  const int i0  = (blockIdx.x / tilesC) * TH; // output tile origin
  const int j0  = (blockIdx.x % tilesC) * TW;

  if (MODE == 2 && tid < 4) lacc[tid] = 0.0f;

  // ---- load reflect-padded input tile into LDS (coalesced) ----
  const size_t plane = (size_t)gy * H * Wsi;
  for (int idx = tid; idx < TR * TC; idx += 256) {
    int r = idx / TC, c = idx - r * TC;
    int gr = refl(2 * i0 + r - 2, H);
    int gc = refl(2 * j0 + c - 2, W);
    size_t g = plane + (size_t)gr * Wsi + gc;
    float v = (MODE == 0) ? (inA[g] - inB[g]) : inA[g];
    tile[r * TSTR + c] = v;
  }
  __syncthreads();

  const int lane = tid & 31;
  const int wid  = tid >> 5;         // 8 waves per block
  const int m    = lane & 15;        // A: M index (j offset); B/D: N index (band)
  const int kk   = (lane >> 4) << 1; // K sub-offset: 0 or 2

  // ---- constant B fragments (branchless, no stack arrays -> no scratch) ----
  // B[k=4c+kk+{0,1}, n=m] = fr[m][c] * fc[m][kk+{0,1}], bands m<4, else 0.
  v2f bfrag[4];
  {
    const bool rHi = (m & 1);        // band row filter: hi for lh, hh
    const bool cHi = (m & 2) != 0;   // band col filter: hi for hl, hh
    const bool up  = (lane >> 4);    // kk == 2
    const float zv = (m < 4) ? 1.0f : 0.0f;
    const float fc0 = (cHi ? (up ? HI2 : HI0) : (up ? LO2 : LO0)) * zv;
    const float fc1 = (cHi ? (up ? HI3 : HI1) : (up ? LO3 : LO1)) * zv;
    const float fr0 = rHi ? HI0 : LO0, fr1 = rHi ? HI1 : LO1;
    const float fr2 = rHi ? HI2 : LO2, fr3 = rHi ? HI3 : LO3;
    bfrag[0].x = fr0 * fc0; bfrag[0].y = fr0 * fc1;
    bfrag[1].x = fr1 * fc0; bfrag[1].y = fr1 * fc1;
    bfrag[2].x = fr2 * fc0; bfrag[2].y = fr2 * fc1;
    bfrag[3].x = fr3 * fc0; bfrag[3].y = fr3 * fc1;
  }

  // Per-lane output base (hoisted 64-bit math; only dereferenced when valid).
  const int jb = j0 + ((lane >> 4) << 3);  // + jg*16 added per chain
  float* const obase =
      (MODE == 2) ? nullptr
                  : out + (size_t)(4 * gy + m) * ((size_t)Hp * Wso) + jb;

  float red = 0.0f;

  // 64 chains (16 rows x 4 col-groups), 8 per wave; each chain: 4 chained WMMAs
#pragma unroll 1
  for (int s = 0; s < 8; ++s) {
    const int chain = wid * 8 + s;
    const int ti = chain >> 2;  // output row in tile
    const int jg = chain & 3;   // 16-col group

    v8f d = {0.f, 0.f, 0.f, 0.f, 0.f, 0.f, 0.f, 0.f};
#pragma unroll
    for (int c = 0; c < 4; ++c) {  // filter row a = c, taps b = kk..kk+1
      const int off = (2 * ti + c) * TSTR + 2 * (jg * 16 + m) + kk;
      v2f a = *(const v2f*)(&tile[off]);
      d = __builtin_amdgcn_wmma_f32_16x16x4_f32(false, a, false, bfrag[c],
                                                (short)0, d, false, false);
    }

    const int i  = i0 + ti;
    const int jc = jg * 16;  // column-group offset on top of jb
    union { v8f v; float f[8]; float4 q[2]; } u; u.v = d;

    if (MODE == 2) {
      if (m < 4 && i < Hp) {
#pragma unroll
        for (int v = 0; v < 8; ++v)
          if (jb + jc + v < Wp) red += u.f[v] * u.f[v];
      }
    } else {
      if (m < 4 && i < Hp && jb + jc < Wp) {
        // 32B-aligned; tail beyond Wp lands in pad columns (never read back).
        float* o = obase + i * Wso + jc;     // 32-bit mad + 64-bit add
        ((float4*)o)[0] = u.q[0];            // global_store_b128
        ((float4*)o)[1] = u.q[1];            // global_store_b128
      }
    }
  }

  if (MODE == 2) {
    if (m < 4) atomicAdd(&lacc[m], red);
    __syncthreads();
    if (tid < 4) atomicAdd(&gacc[(gy % nb) * 4 + tid], lacc[tid]);
  }
}

__global__ void zero_acc(float* gacc) { gacc[threadIdx.x] = 0.0f; }

__global__ void finalize(const float* __restrict__ gacc, float* __restrict__ out) {
  __shared__ float s[64];
  int t = threadIdx.x;
  s[t] = gacc[t] * (t == 0 ? 0.5f : 1.0f);  // leaf-0 weight 0.5
  for (int off = 32; off > 0; off >>= 1) {
    __syncthreads();
    if (t < off) s[t] += s[t + off];
  }
  __syncthreads();
  if (t == 0) out[0] = s[0] * (1.0f / 836352.0f);  // mean over 192*66*66
}

extern "C" void kernel_launch(void* const* d_in, const int* in_sizes, int n_in,
                              void* d_out, int out_size, void* d_ws, size_t ws_size,
                              hipStream_t stream) {
  (void)in_sizes; (void)n_in; (void)out_size; (void)ws_size;
  const float* preds = (const float*)d_in[0];  // [192, 512, 512] flat
  const float* targs = (const float*)d_in[1];

  // Padded row strides: multiples of 8 floats (32B) >= valid width.
  const int W1 = 264;  // level-1 output rows: 257 valid cols
  const int W2 = 136;  // level-2 output rows: 130 valid cols

  // 8 chunks of 24 planes; workspace ~53 MB
  const int CH = 24;
  float* buf1 = (float*)d_ws;                      // [24*4, 257, 264]
  float* buf2 = buf1 + (size_t)CH * 4 * 257 * W1;  // [24*16, 130, 136]
  float* gacc = buf2 + (size_t)CH * 16 * 130 * W2; // [64]

  zero_acc<<<1, 64, 0, stream>>>(gacc);

  for (int c = 0; c < 8; ++c) {
    const size_t base = (size_t)c * CH * 512 * 512;
    // L1: 512 -> 257 (diff fused), 1 band-group in, 4 out
    dwt_kernel<0><<<dim3(17 * 5, CH), 256, 0, stream>>>(
        preds + base, targs + base, buf1, nullptr,
        512, 512, 512, 257, 257, W1, 1, 5);
    // L2: 257 -> 130, 4 in, 16 out
    dwt_kernel<1><<<dim3(9 * 3, CH * 4), 256, 0, stream>>>(
        buf1, nullptr, buf2, nullptr,
        257, 257, W1, 130, 130, W2, 4, 3);
    // L3: 130 -> 66, 16 in, 64 leaves, fused sum-of-squares reduction
    dwt_kernel<2><<<dim3(5 * 2, CH * 16), 256, 0, stream>>>(
        buf2, nullptr, nullptr, gacc,
        130, 130, W2, 66, 66, 0, 16, 2);
  }

  finalize<<<1, 64, 0, stream>>>(gacc, (float*)d_out);
}